// ST_GAT_21612275434309
// MI455X (gfx1250) — compile-verified
//
#include <hip/hip_runtime.h>
#include <hip/hip_bf16.h>
#include <math.h>

#define H 8
#define F 12
#define N_NODES 1024
#define BATCH 32
#define NTOT (BATCH * N_NODES)   /* 32768 */
#define N_PRED 9
#define HF (H * F)               /* 96 */

typedef __attribute__((ext_vector_type(2))) float v2f;
typedef __attribute__((ext_vector_type(8))) float v8f;

// ---- V_WMMA_F32_16X16X4_F32 wrapper (wave32) ------------------------------
__device__ __forceinline__ v8f wmma4(v2f a, v2f b, v8f c) {
    // 8 args: (neg_a, A, neg_b, B, c_mod, C, reuse_a, reuse_b)
    return __builtin_amdgcn_wmma_f32_16x16x4_f32(false, a, false, b,
                                                 (short)0, c, false, false);
}

// Order-preserving float <-> uint encoding for atomic float-max via atomicMax(u32)
__device__ __forceinline__ unsigned enc_f32(float f) {
    unsigned b = __float_as_uint(f);
    return (b & 0x80000000u) ? ~b : (b | 0x80000000u);
}
__device__ __forceinline__ float dec_f32(unsigned e) {
    unsigned b = (e & 0x80000000u) ? (e & 0x7FFFFFFFu) : ~e;
    return __uint_as_float(b);
}
#define ENC_NEG_INF 0x007FFFFFu   /* enc(-inf) */

// ===========================================================================
// Stage 1: z = feat @ fc_w   [32768,12] x [12,96] -> [32768,96]
// 2048 M-tiles x 6 N-tiles = 12288 tiles; 8 waves/block -> 1536 blocks.
// ===========================================================================
__global__ void k_zgemm(const float* __restrict__ feat,
                        const float* __restrict__ fcw,
                        float* __restrict__ z) {
    const int wave = threadIdx.x >> 5, lane = threadIdx.x & 31;
    const int lo = lane & 15, hi = lane >> 4;
    const int tile = blockIdx.x * 8 + wave;
    const int m0 = (tile / 6) * 16, n0 = (tile % 6) * 16;
    v8f c = {};
    for (int k0 = 0; k0 < F; k0 += 4) {
        v2f a, b;
        const int ka = k0 + 2 * hi;
        a.x = feat[(m0 + lo) * F + ka];
        a.y = feat[(m0 + lo) * F + ka + 1];
        b.x = fcw[(ka) * HF + n0 + lo];
        b.y = fcw[(ka + 1) * HF + n0 + lo];
        c = wmma4(a, b, c);
    }
#pragma unroll
    for (int v = 0; v < 8; v++)
        z[(m0 + v + 8 * hi) * HF + n0 + lo] = c[v];
}

// ===========================================================================
// Stage 2: el/er = einsum(z, attn_l/r), init segment-max/denom accumulators
// ===========================================================================
__global__ void k_elr(const float* __restrict__ z,
                      const float* __restrict__ al, const float* __restrict__ ar,
                      float* __restrict__ el, float* __restrict__ er,
                      unsigned* __restrict__ menc, float* __restrict__ denom) {
    const int i = blockIdx.x * blockDim.x + threadIdx.x;
    if (i >= NTOT * H) return;
    const int n = i >> 3, h = i & 7;
    const float* zp = z + n * HF + h * F;
    float sl = 0.f, sr = 0.f;
#pragma unroll
    for (int f = 0; f < F; f++) {
        sl += zp[f] * al[h * F + f];
        sr += zp[f] * ar[h * F + f];
    }
    el[i] = sl; er[i] = sr;
    menc[i] = ENC_NEG_INF;
    denom[i] = 0.f;
}

__global__ void k_zero(float* __restrict__ p, int n) {
    const int i = blockIdx.x * blockDim.x + threadIdx.x;
    if (i < n) p[i] = 0.f;
}

// ===========================================================================
// Stage 3a: edge-softmax pass 1 -> segment max per (dst, head) via u32 atomicMax
// ===========================================================================
__global__ void k_edge_max(const int* __restrict__ src, const int* __restrict__ dst,
                           const float* __restrict__ el, const float* __restrict__ er,
                           unsigned* __restrict__ menc, int E) {
    const int e = blockIdx.x * blockDim.x + threadIdx.x;
    if (e >= E) return;
    const int s = src[e], d = dst[e];
#pragma unroll
    for (int h = 0; h < H; h++) {
        float v = el[s * H + h] + er[d * H + h];
        v = v > 0.f ? v : 0.2f * v;
        atomicMax(&menc[d * H + h], enc_f32(v));
    }
}

// Stage 3b: pass 2 -> denom = segment_sum(exp(e - m))
__global__ void k_edge_sum(const int* __restrict__ src, const int* __restrict__ dst,
                           const float* __restrict__ el, const float* __restrict__ er,
                           const unsigned* __restrict__ menc,
                           float* __restrict__ denom, int E) {
    const int e = blockIdx.x * blockDim.x + threadIdx.x;
    if (e >= E) return;
    const int s = src[e], d = dst[e];
#pragma unroll
    for (int h = 0; h < H; h++) {
        float v = el[s * H + h] + er[d * H + h];
        v = v > 0.f ? v : 0.2f * v;
        atomicAdd(&denom[d * H + h], expf(v - dec_f32(menc[d * H + h])));
    }
}

// Stage 3c: pass 3 -> alpha out + h_accum[dst] += alpha * z[src]  (per edge,head)
__global__ void k_edge_aggr(const int* __restrict__ src, const int* __restrict__ dst,
                            const float* __restrict__ el, const float* __restrict__ er,
                            const unsigned* __restrict__ menc,
                            const float* __restrict__ denom,
                            const float* __restrict__ z,
                            float* __restrict__ haccum,
                            float* __restrict__ attn_out, int E) {
    const int i = blockIdx.x * blockDim.x + threadIdx.x;
    if (i >= E * H) return;
    const int e = i >> 3, h = i & 7;
    const int s = src[e], d = dst[e];
    float v = el[s * H + h] + er[d * H + h];
    v = v > 0.f ? v : 0.2f * v;
    const float alpha = expf(v - dec_f32(menc[d * H + h])) / denom[d * H + h];
    attn_out[i] = alpha;
    const float* zs = z + s * HF + h * F;
    float* ha = haccum + d * HF + h * F;
#pragma unroll
    for (int f = 0; f < F; f++)
        atomicAdd(&ha[f], alpha * zs[f]);
}

// ===========================================================================
// Stage 4: head-mean(+bias) and repack into LSTM layout
// Xall[(t*32+b)*1024 + node] = mean_h(haccum[n,h,t] + bias[h,t]), n = b*1024+node
// ===========================================================================
__global__ void k_mean(const float* __restrict__ haccum,
                       const float* __restrict__ bias,
                       float* __restrict__ Xall) {
    const int i = blockIdx.x * blockDim.x + threadIdx.x;
    if (i >= NTOT * F) return;
    const int n = i / F, t = i % F;
    float s = 0.f;
#pragma unroll
    for (int h = 0; h < H; h++)
        s += haccum[n * HF + h * F + t] + bias[h * F + t];
    s *= (1.0f / (float)H);
    const int b = n >> 10, node = n & (N_NODES - 1);
    Xall[(t * BATCH + b) * N_NODES + node] = s;
}

// ===========================================================================
// Stage 5: LSTM1 input projection: g1 = Xall @ w_ih1^T + b_ih1 + b_hh1
// [384,1024] x [1024,128]; 24x8 = 192 tiles -> 24 blocks x 8 waves; K=1024
// ===========================================================================
__global__ void k_gates1(const float* __restrict__ Xall,
                         const float* __restrict__ w_ih1,
                         const float* __restrict__ b_ih1,
                         const float* __restrict__ b_hh1,
                         float* __restrict__ g1) {
    const int wave = threadIdx.x >> 5, lane = threadIdx.x & 31;
    const int lo = lane & 15, hi = lane >> 4;
    const int tile = blockIdx.x * 8 + wave;
    const int m0 = (tile >> 3) * 16, n0 = (tile & 7) * 16;
    v8f c = {};
    for (int k0 = 0; k0 < N_NODES; k0 += 4) {
        v2f a, b;
        const int ka = k0 + 2 * hi;
        a.x = Xall[(m0 + lo) * N_NODES + ka];
        a.y = Xall[(m0 + lo) * N_NODES + ka + 1];
        b.x = w_ih1[(n0 + lo) * N_NODES + ka];       // B = w_ih1^T
        b.y = w_ih1[(n0 + lo) * N_NODES + ka + 1];
        c = wmma4(a, b, c);
    }
    const float bias = b_ih1[n0 + lo] + b_hh1[n0 + lo];
#pragma unroll
    for (int v = 0; v < 8; v++)
        g1[(m0 + v + 8 * hi) * 128 + n0 + lo] = c[v] + bias;
}

// ===========================================================================
// Stage 6: LSTM1 recurrence (hidden=32, batch=32, 12 steps) — one workgroup
// ===========================================================================
__global__ void k_lstm1(const float* __restrict__ g1,
                        const float* __restrict__ w_hh1,
                        float* __restrict__ xs2) {
    __shared__ float hs[BATCH * 32];
    __shared__ float cs[BATCH * 32];
    __shared__ float w[128 * 32];
    const int tid = threadIdx.x;                 // 1024 threads
    for (int i = tid; i < 128 * 32; i += 1024) w[i] = w_hh1[i];
    hs[tid] = 0.f; cs[tid] = 0.f;
    __syncthreads();
    const int b = tid >> 5, u = tid & 31;
    for (int t = 0; t < F; t++) {
        const float* gr = g1 + (t * BATCH + b) * 128;
        float xi = gr[u], xf = gr[32 + u], xg = gr[64 + u], xo = gr[96 + u];
        const float* hb = hs + b * 32;
#pragma unroll
        for (int k = 0; k < 32; k++) {
            const float hk = hb[k];
            xi += hk * w[u * 32 + k];
            xf += hk * w[(32 + u) * 32 + k];
            xg += hk * w[(64 + u) * 32 + k];
            xo += hk * w[(96 + u) * 32 + k];
        }
        const float ig = 1.f / (1.f + expf(-xi));
        const float fg = 1.f / (1.f + expf(-xf));
        const float gg = tanhf(xg);
        const float og = 1.f / (1.f + expf(-xo));
        const float cn = fg * cs[b * 32 + u] + ig * gg;
        const float hn = og * tanhf(cn);
        __syncthreads();
        hs[b * 32 + u] = hn; cs[b * 32 + u] = cn;
        xs2[(t * BATCH + b) * 32 + u] = hn;
        __syncthreads();
    }
}

// ===========================================================================
// Stage 7: LSTM2 input projection: g2 = xs2 @ w_ih2^T + b_ih2 + b_hh2
// [384,32] x [32,512]; 24x32 = 768 tiles -> 96 blocks x 8 waves; K=32
// ===========================================================================
__global__ void k_gates2(const float* __restrict__ xs2,
                         const float* __restrict__ w_ih2,
                         const float* __restrict__ b_ih2,
                         const float* __restrict__ b_hh2,
                         float* __restrict__ g2) {
    const int wave = threadIdx.x >> 5, lane = threadIdx.x & 31;
    const int lo = lane & 15, hi = lane >> 4;
    const int tile = blockIdx.x * 8 + wave;
    const int m0 = (tile >> 5) * 16, n0 = (tile & 31) * 16;
    v8f c = {};
    for (int k0 = 0; k0 < 32; k0 += 4) {
        v2f a, b;
        const int ka = k0 + 2 * hi;
        a.x = xs2[(m0 + lo) * 32 + ka];
        a.y = xs2[(m0 + lo) * 32 + ka + 1];
        b.x = w_ih2[(n0 + lo) * 32 + ka];            // B = w_ih2^T
        b.y = w_ih2[(n0 + lo) * 32 + ka + 1];
        c = wmma4(a, b, c);
    }
    const float bias = b_ih2[n0 + lo] + b_hh2[n0 + lo];
#pragma unroll
    for (int v = 0; v < 8; v++)
        g2[(m0 + v + 8 * hi) * 512 + n0 + lo] = c[v] + bias;
}

// ===========================================================================
// Stage 8: LSTM2 recurrence (hidden=128, batch=32, 12 steps) — one workgroup
// w_hh2 (256KB) streamed from L2. Only last h is needed downstream.
// ===========================================================================
__global__ void k_lstm2(const float* __restrict__ g2,
                        const float* __restrict__ w_hh2,
                        float* __restrict__ last) {
    __shared__ float hs[BATCH * 128];
    __shared__ float cs[BATCH * 128];
    const int tid = threadIdx.x;                 // 1024 threads; 4 (b,u) pairs each
    for (int i = tid; i < BATCH * 128; i += 1024) { hs[i] = 0.f; cs[i] = 0.f; }
    __syncthreads();
    for (int t = 0; t < F; t++) {
        float hn[4], cn[4];
#pragma unroll
        for (int p = 0; p < 4; p++) {
            const int pair = tid + p * 1024;
            const int b = pair >> 7, u = pair & 127;
            const float* gr = g2 + (t * BATCH + b) * 512;
            float xi = gr[u], xf = gr[128 + u], xg = gr[256 + u], xo = gr[384 + u];
            const float* hb = hs + b * 128;
            for (int k = 0; k < 128; k++) {
                const float hk = hb[k];
                xi += hk * w_hh2[u * 128 + k];
                xf += hk * w_hh2[(128 + u) * 128 + k];
                xg += hk * w_hh2[(256 + u) * 128 + k];
                xo += hk * w_hh2[(384 + u) * 128 + k];
            }
            const float ig = 1.f / (1.f + expf(-xi));
            const float fg = 1.f / (1.f + expf(-xf));
            const float gg = tanhf(xg);
            const float og = 1.f / (1.f + expf(-xo));
            cn[p] = fg * cs[pair] + ig * gg;
            hn[p] = og * tanhf(cn[p]);
        }
        __syncthreads();
#pragma unroll
        for (int p = 0; p < 4; p++) {
            const int pair = tid + p * 1024;
            hs[pair] = hn[p]; cs[pair] = cn[p];
        }
        __syncthreads();
    }
    for (int p = 0; p < 4; p++) {
        const int pair = tid + p * 1024;
        last[pair] = hs[pair];
    }
}

// ===========================================================================
// Stage 9: out = last @ lin_w^T + lin_b   [32,128] x [128,9216]
// 2x576 = 1152 tiles -> 144 blocks x 8 waves; K=128
// ===========================================================================
__global__ void k_linear(const float* __restrict__ last,
                         const float* __restrict__ lin_w,
                         const float* __restrict__ lin_b,
                         float* __restrict__ out) {
    const int wave = threadIdx.x >> 5, lane = threadIdx.x & 31;
    const int lo = lane & 15, hi = lane >> 4;
    const int tile = blockIdx.x * 8 + wave;
    const int m0 = (tile & 1) * 16, n0 = (tile >> 1) * 16;
    v8f c = {};
    for (int k0 = 0; k0 < 128; k0 += 4) {
        v2f a, b;
        const int ka = k0 + 2 * hi;
        a.x = last[(m0 + lo) * 128 + ka];
        a.y = last[(m0 + lo) * 128 + ka + 1];
        b.x = lin_w[(n0 + lo) * 128 + ka];           // B = lin_w^T
        b.y = lin_w[(n0 + lo) * 128 + ka + 1];
        c = wmma4(a, b, c);
    }
    const float bias = lin_b[n0 + lo];
#pragma unroll
    for (int v = 0; v < 8; v++)
        out[(m0 + v + 8 * hi) * (N_NODES * N_PRED) + n0 + lo] = c[v] + bias;
}

// ===========================================================================
extern "C" void kernel_launch(void* const* d_in, const int* in_sizes, int n_in,
                              void* d_out, int out_size, void* d_ws, size_t ws_size,
                              hipStream_t stream) {
    const float* feat   = (const float*)d_in[0];
    const int*   esrc   = (const int*)  d_in[1];
    const int*   edst   = (const int*)  d_in[2];
    const float* fc_w   = (const float*)d_in[3];
    const float* attn_l = (const float*)d_in[4];
    const float* attn_r = (const float*)d_in[5];
    const float* gbias  = (const float*)d_in[6];
    const float* w_ih1  = (const float*)d_in[7];
    const float* w_hh1  = (const float*)d_in[8];
    const float* b_ih1  = (const float*)d_in[9];
    const float* b_hh1  = (const float*)d_in[10];
    const float* w_ih2  = (const float*)d_in[11];
    const float* w_hh2  = (const float*)d_in[12];
    const float* b_ih2  = (const float*)d_in[13];
    const float* b_hh2  = (const float*)d_in[14];
    const float* lin_w  = (const float*)d_in[15];
    const float* lin_b  = (const float*)d_in[16];
    const int E = in_sizes[1];
    (void)n_in; (void)out_size; (void)ws_size;

    float* out  = (float*)d_out;                  // [32, 9216]
    float* attn = out + NTOT * N_PRED;            // [E, H, 1]

    // workspace carve-out (256B aligned regions)
    char* wsb = (char*)d_ws;
    size_t off = 0;
    auto carve = [&](size_t nbytes) -> void* {
        void* p = wsb + off;
        off = (off + nbytes + 255) & ~(size_t)255;
        return p;
    };
    float*    z      = (float*)   carve(sizeof(float) * NTOT * HF);
    float*    el     = (float*)   carve(sizeof(float) * NTOT * H);
    float*    er     = (float*)   carve(sizeof(float) * NTOT * H);
    unsigned* menc   = (unsigned*)carve(sizeof(unsigned) * NTOT * H);
    float*    denom  = (float*)   carve(sizeof(float) * NTOT * H);
    float*    haccum = (float*)   carve(sizeof(float) * NTOT * HF);
    float*    Xall   = (float*)   carve(sizeof(float) * F * BATCH * N_NODES);
    float*    g1     = (float*)   carve(sizeof(float) * F * BATCH * 128);
    float*    xs2    = (float*)   carve(sizeof(float) * F * BATCH * 32);
    float*    g2     = (float*)   carve(sizeof(float) * F * BATCH * 512);
    float*    last   = (float*)   carve(sizeof(float) * BATCH * 128);

    k_zgemm<<<1536, 256, 0, stream>>>(feat, fc_w, z);
    k_elr<<<(NTOT * H + 255) / 256, 256, 0, stream>>>(z, attn_l, attn_r, el, er, menc, denom);
    k_zero<<<(NTOT * HF + 255) / 256, 256, 0, stream>>>(haccum, NTOT * HF);
    k_edge_max<<<(E + 255) / 256, 256, 0, stream>>>(esrc, edst, el, er, menc, E);
    k_edge_sum<<<(E + 255) / 256, 256, 0, stream>>>(esrc, edst, el, er, menc, denom, E);
    k_edge_aggr<<<(E * H + 255) / 256, 256, 0, stream>>>(esrc, edst, el, er, menc, denom, z,
                                                         haccum, attn, E);
    k_mean<<<(NTOT * F + 255) / 256, 256, 0, stream>>>(haccum, gbias, Xall);
    k_gates1<<<24, 256, 0, stream>>>(Xall, w_ih1, b_ih1, b_hh1, g1);
    k_lstm1<<<1, 1024, 0, stream>>>(g1, w_hh1, xs2);
    k_gates2<<<96, 256, 0, stream>>>(xs2, w_ih2, b_ih2, b_hh2, g2);
    k_lstm2<<<1, 1024, 0, stream>>>(g2, w_hh2, last);
    k_linear<<<144, 256, 0, stream>>>(last, lin_w, lin_b, out);
}